// GraphProcessor_6262062317902
// MI455X (gfx1250) — compile-verified
//
#include <hip/hip_runtime.h>
#include <math.h>

#define NNODES 100000
#define NEDGES 1600000
#define EPLUS  (NNODES + NEDGES)   // edges + self loops = 1,700,000
#define FEAT   128                 // H*C
#define HEADS  2
#define CHAN   64
#define NGRAPH 64
#define NEG_SLOPE 0.2f

typedef float v2f __attribute__((ext_vector_type(2)));
typedef float v8f __attribute__((ext_vector_type(8)));

// ---------------------------------------------------------------------------
// GEMM: C[N,128] = A[N,128] * W[128,128] using V_WMMA_F32_16X16X4_F32.
// Block = 256 threads (8 waves). Block owns a 16-row strip; wave w owns the
// 16x16 tile at columns [16w, 16w+16). A strip staged in LDS (padded).
// Fragment layouts per CDNA5 ISA 7.12.2:
//   A 16x4 f32 : lane = {m = lane&15}, vgpr0/1 = K = 2*(lane>>4) + {0,1}
//   B 4x16 f32 : lane = {n = lane&15}, vgpr0/1 = K = 2*(lane>>4) + {0,1}
//   C 16x16 f32: vgpr i -> row i + 8*(lane>>4), col = lane&15
// ---------------------------------------------------------------------------
__global__ __launch_bounds__(256) void gemm_kernel(
    const float* __restrict__ A, const float* __restrict__ W,
    float* __restrict__ C)
{
  __shared__ float sA[16 * 132];            // stride 132 -> conflict-free col reads
  const int t    = threadIdx.x;
  const int wave = t >> 5;
  const int lane = t & 31;
  const int mbase = blockIdx.x * 16;

  // stage 16x128 strip of A (coalesced float4)
  {
    const float4* A4 = (const float4*)(A + (size_t)mbase * FEAT);
    for (int i = t; i < 16 * (FEAT / 4); i += 256) {
      int r  = i >> 5;                      // 32 float4 per row
      int c4 = i & 31;
      float4 v = A4[r * 32 + c4];
      float* dp = &sA[r * 132 + c4 * 4];
      dp[0] = v.x; dp[1] = v.y; dp[2] = v.z; dp[3] = v.w;
    }
  }
  __syncthreads();

  const int nbase = wave * 16;
  const int lidx  = lane & 15;              // m-row of A frag / n-col of B frag
  const int kk    = (lane >> 4) << 1;       // 0 or 2
  const float* sa = &sA[lidx * 132];

  v8f acc = {0.f, 0.f, 0.f, 0.f, 0.f, 0.f, 0.f, 0.f};
#pragma unroll 4
  for (int k0 = 0; k0 < FEAT; k0 += 4) {
    v2f a, b;
    a[0] = sa[k0 + kk];
    a[1] = sa[k0 + kk + 1];
    b[0] = W[(k0 + kk) * FEAT + nbase + lidx];
    b[1] = W[(k0 + kk + 1) * FEAT + nbase + lidx];
    acc = __builtin_amdgcn_wmma_f32_16x16x4_f32(false, a, false, b,
                                                (short)0, acc, false, false);
  }

  const int col  = nbase + lidx;
  const int rofs = (lane >> 4) << 3;        // 0 or 8
#pragma unroll
  for (int i = 0; i < 8; ++i)
    C[(size_t)(mbase + rofs + i) * FEAT + col] = acc[i];
}

// ---------------------------------------------------------------------------
// Per-node attention logits: alpha_src/dst[n,h] = dot(h[n,h,:], a_src/dst[h,:])
// ---------------------------------------------------------------------------
__global__ void alpha_kernel(const float* __restrict__ h,
                             const float* __restrict__ w_src,
                             const float* __restrict__ w_dst,
                             float* __restrict__ asrc,
                             float* __restrict__ adst)
{
  int t = blockIdx.x * blockDim.x + threadIdx.x;
  if (t >= NNODES * HEADS) return;
  int n = t >> 1, hd = t & 1;
  const float* hp = h + (size_t)n * FEAT + hd * CHAN;
  const float* ws = w_src + hd * CHAN;
  const float* wd = w_dst + hd * CHAN;
  float s = 0.f, d = 0.f;
#pragma unroll 8
  for (int c = 0; c < CHAN; ++c) { s += hp[c] * ws[c]; d += hp[c] * wd[c]; }
  asrc[t] = s;
  adst[t] = d;
}

__global__ void fill_kernel(float* __restrict__ p, float v, int n)
{
  int t = blockIdx.x * blockDim.x + threadIdx.x;
  if (t < n) p[t] = v;
}

// sign-split float atomic max (init must be -inf)
__device__ __forceinline__ void atomicMaxF(float* addr, float v)
{
  if (v >= 0.f) atomicMax((int*)addr, __float_as_int(v));
  else          atomicMin((unsigned int*)addr, __float_as_uint(v));
}

__device__ __forceinline__ void edge_sd(int e, int& s, int& d,
                                        const int* __restrict__ ei)
{
  if (e < NEDGES) { s = ei[e]; d = ei[NEDGES + e]; }
  else            { s = d = e - NEDGES; }           // self loops
}

__global__ void edge_max_kernel(const int* __restrict__ ei,
                                const float* __restrict__ asrc,
                                const float* __restrict__ adst,
                                float* __restrict__ m)
{
  int e = blockIdx.x * blockDim.x + threadIdx.x;
  if (e >= EPLUS) return;
  int s, d; edge_sd(e, s, d, ei);
#pragma unroll
  for (int hd = 0; hd < HEADS; ++hd) {
    float v = asrc[s * 2 + hd] + adst[d * 2 + hd];
    v = v > 0.f ? v : NEG_SLOPE * v;                // leaky relu
    atomicMaxF(&m[d * 2 + hd], v);
  }
}

__global__ void edge_exp_kernel(const int* __restrict__ ei,
                                const float* __restrict__ asrc,
                                const float* __restrict__ adst,
                                const float* __restrict__ m,
                                float* __restrict__ exbuf,
                                float* __restrict__ denom)
{
  int e = blockIdx.x * blockDim.x + threadIdx.x;
  if (e >= EPLUS) return;
  int s, d; edge_sd(e, s, d, ei);
#pragma unroll
  for (int hd = 0; hd < HEADS; ++hd) {
    float v = asrc[s * 2 + hd] + adst[d * 2 + hd];
    v = v > 0.f ? v : NEG_SLOPE * v;
    float ex = expf(v - m[d * 2 + hd]);
    exbuf[e * 2 + hd] = ex;
    atomicAdd(&denom[d * 2 + hd], ex);
  }
}

// one thread per (edge, feature): out[dst,f] += alpha * h[src,f]
__global__ void edge_agg_kernel(const int* __restrict__ ei,
                                const float* __restrict__ exbuf,
                                const float* __restrict__ denom,
                                const float* __restrict__ h,
                                float* __restrict__ out)
{
  int t = blockIdx.x * blockDim.x + threadIdx.x;    // EPLUS*128 threads
  int e = t >> 7;
  if (e >= EPLUS) return;
  int f  = t & 127;
  int hd = f >> 6;
  int s, d; edge_sd(e, s, d, ei);
  float coef = exbuf[e * 2 + hd] / (denom[d * 2 + hd] + 1e-16f);
  atomicAdd(&out[(size_t)d * FEAT + f], coef * h[(size_t)s * FEAT + f]);
}

__global__ void bias_relu_kernel(float* __restrict__ o,
                                 const float* __restrict__ b)
{
  int t = blockIdx.x * blockDim.x + threadIdx.x;
  if (t >= NNODES * FEAT) return;
  float v = o[t] + b[t & 127];
  o[t] = v > 0.f ? v : 0.f;
}

__global__ void count_kernel(const int* __restrict__ batch,
                             int* __restrict__ counts)
{
  int n = blockIdx.x * blockDim.x + threadIdx.x;
  if (n >= NNODES) return;
  atomicAdd(&counts[batch[n]], 1);
}

__global__ void pool_acc_kernel(const int* __restrict__ batch,
                                const float* __restrict__ o,
                                float* __restrict__ pooled)
{
  int t = blockIdx.x * blockDim.x + threadIdx.x;
  if (t >= NNODES * FEAT) return;
  int n = t >> 7, f = t & 127;
  atomicAdd(&pooled[batch[n] * FEAT + f], o[t]);
}

__global__ void pool_fin_kernel(float* __restrict__ pooled,
                                const int* __restrict__ counts,
                                const float* __restrict__ b2)
{
  int t = blockIdx.x * blockDim.x + threadIdx.x;
  if (t >= NGRAPH * FEAT) return;
  int g = t >> 7, f = t & 127;
  int c = counts[g];
  pooled[t] = (c > 0) ? (pooled[t] / (float)c + b2[f]) : 0.f;
}

// ---------------------------------------------------------------------------
extern "C" void kernel_launch(void* const* d_in, const int* in_sizes, int n_in,
                              void* d_out, int out_size, void* d_ws, size_t ws_size,
                              hipStream_t stream)
{
  (void)in_sizes; (void)n_in; (void)out_size; (void)ws_size;

  const float* x     = (const float*)d_in[0];
  const int*   ei    = (const int*)  d_in[1];
  const int*   batch = (const int*)  d_in[2];
  const float* W1    = (const float*)d_in[3];
  const float* a1s   = (const float*)d_in[4];
  const float* a1d   = (const float*)d_in[5];
  const float* b1    = (const float*)d_in[6];
  const float* W2    = (const float*)d_in[7];
  const float* a2s   = (const float*)d_in[8];
  const float* a2d   = (const float*)d_in[9];
  const float* b2    = (const float*)d_in[10];
  float* out = (float*)d_out;

  // workspace layout
  float* ws   = (float*)d_ws;
  float* bufH  = ws;  ws += (size_t)NNODES * FEAT;   // transformed features h
  float* bufO  = ws;  ws += (size_t)NNODES * FEAT;   // aggregated output
  float* asrc  = ws;  ws += NNODES * HEADS;
  float* adst  = ws;  ws += NNODES * HEADS;
  float* mbuf  = ws;  ws += NNODES * HEADS;
  float* dbuf  = ws;  ws += NNODES * HEADS;
  float* exbuf = ws;  ws += (size_t)EPLUS * HEADS;
  int*   counts = (int*)ws;

  const int TB = 256;
  auto cdiv = [](long long a, long long b) { return (int)((a + b - 1) / b); };

  for (int layer = 0; layer < 2; ++layer) {
    const float* in   = (layer == 0) ? x   : bufO;
    const float* W    = (layer == 0) ? W1  : W2;
    const float* a_s  = (layer == 0) ? a1s : a2s;
    const float* a_d  = (layer == 0) ? a1d : a2d;

    gemm_kernel<<<NNODES / 16, 256, 0, stream>>>(in, W, bufH);
    alpha_kernel<<<cdiv(NNODES * HEADS, TB), TB, 0, stream>>>(bufH, a_s, a_d, asrc, adst);
    fill_kernel<<<cdiv(NNODES * HEADS, TB), TB, 0, stream>>>(mbuf, -INFINITY, NNODES * HEADS);
    fill_kernel<<<cdiv(NNODES * HEADS, TB), TB, 0, stream>>>(dbuf, 0.f, NNODES * HEADS);
    edge_max_kernel<<<cdiv(EPLUS, TB), TB, 0, stream>>>(ei, asrc, adst, mbuf);
    edge_exp_kernel<<<cdiv(EPLUS, TB), TB, 0, stream>>>(ei, asrc, adst, mbuf, exbuf, dbuf);
    fill_kernel<<<cdiv((long long)NNODES * FEAT, TB), TB, 0, stream>>>(bufO, 0.f, NNODES * FEAT);
    edge_agg_kernel<<<cdiv((long long)EPLUS * FEAT, TB), TB, 0, stream>>>(ei, exbuf, dbuf, bufH, bufO);
    if (layer == 0)
      bias_relu_kernel<<<cdiv((long long)NNODES * FEAT, TB), TB, 0, stream>>>(bufO, b1);
    // layer 1 result (relu(out+b1)) now in bufO -> input of layer 2
    // layer 2 bias folded into pool finalize
  }

  // global mean pool over batch ids (+ b2)
  fill_kernel<<<cdiv(NGRAPH * FEAT, TB), TB, 0, stream>>>(out, 0.f, NGRAPH * FEAT);
  fill_kernel<<<1, TB, 0, stream>>>((float*)counts, 0.f, NGRAPH);
  count_kernel<<<cdiv(NNODES, TB), TB, 0, stream>>>(batch, counts);
  pool_acc_kernel<<<cdiv((long long)NNODES * FEAT, TB), TB, 0, stream>>>(batch, bufO, out);
  pool_fin_kernel<<<cdiv(NGRAPH * FEAT, TB), TB, 0, stream>>>(out, counts, b2);
}